// RateAdaptionEncoder_54589034332166
// MI455X (gfx1250) — compile-verified
//
#include <hip/hip_runtime.h>

typedef float v2f __attribute__((ext_vector_type(2)));
typedef float v8f __attribute__((ext_vector_type(8)));

#define NTOK 4096   // B*H*W
#define CDIM 256
#define RDIM 256
#define NGRP 8
#define LHW  256    // H*W

// ---------------------------------------------------------------------------
// Pass 1: bucket tokens by their rate index (single block, LDS atomics).
// Positions within a group are order-nondeterministic, but each token's final
// output is computed identically regardless of its slot, so d_out is
// deterministic.
// ---------------------------------------------------------------------------
__global__ __launch_bounds__(256) void build_group_lists(
    const int* __restrict__ indexes,
    int* __restrict__ counts,
    unsigned short* __restrict__ lists) {
  __shared__ int scnt[NGRP];
  const int tid = threadIdx.x;
  if (tid < NGRP) scnt[tid] = 0;
  __syncthreads();
  for (int t = tid; t < NTOK; t += 256) {
    int g = indexes[t] & 7;
    int pos = atomicAdd(&scnt[g], 1);          // ds_add_rtn_u32
    lists[g * NTOK + pos] = (unsigned short)t;
  }
  __syncthreads();
  if (tid < NGRP) counts[tid] = scnt[tid];
}

// ---------------------------------------------------------------------------
// Pass 2: per-group dense GEMM tiles via V_WMMA_F32_16X16X4_F32.
// Block = 8 waves; block covers one 16-token tile of one group; wave w covers
// output columns [32w, 32w+32) as two 16-wide WMMA tiles sharing A fragments.
// ---------------------------------------------------------------------------
__global__ __launch_bounds__(256) void rate_adaption_wmma(
    const float* __restrict__ x,        // [B, C, H, W]
    const float* __restrict__ weight,   // [8, C, R]
    const float* __restrict__ bias,     // [8, R]
    const int* __restrict__ rate_choice,// [8]
    const int* __restrict__ counts,     // [8]
    const unsigned short* __restrict__ lists, // [8, 4096]
    float* __restrict__ outY,           // [B, R, H, W]
    float* __restrict__ outM) {         // [B, R, H, W]
  const int g    = blockIdx.x & 7;
  const int tile = blockIdx.x >> 3;
  const int cnt  = counts[g];
  const int t0   = tile * 16;
  if (t0 >= cnt) return;                 // whole block exits: EXEC stays full
  const int nvalid = (cnt - t0 < 16) ? (cnt - t0) : 16;

  __shared__ int s_tok[16];
  const int tid = threadIdx.x;
  if (tid < 16) {
    int i = t0 + tid;
    if (i >= cnt) i = t0;                // clamp; garbage rows masked at store
    s_tok[tid] = (int)lists[g * NTOK + i];
  }
  __syncthreads();

  const int lane  = tid & 31;
  const int wave  = tid >> 5;
  const int half  = lane >> 4;           // 0 => lanes 0-15, 1 => lanes 16-31
  const int lrow  = lane & 15;           // M for A, N for B/C/D
  const int kbase = half * 2;            // K sub-offset within a K=4 step

  // A operand base: this lane's token row, x[b, k, l] = x[b*65536 + k*256 + l]
  const int tokA = s_tok[lrow];
  const int bA = tokA >> 8;
  const int lA = tokA & (LHW - 1);
  const float* aptr = x + (size_t)bA * (CDIM * LHW) + (size_t)kbase * LHW + lA;

  // B operand bases: weight[g, k, r] = weight[g*65536 + k*256 + r]
  const int col0 = wave * 32;
  const float* bptr0 = weight + (size_t)g * (CDIM * RDIM)
                               + (size_t)kbase * RDIM + col0 + lrow;
  const float* bptr1 = bptr0 + 16;

  v8f acc0 = {};
  v8f acc1 = {};
  #pragma unroll 4
  for (int k0 = 0; k0 < CDIM; k0 += 4) {
    v2f a, b0, b1;
    const float* ap  = aptr  + (size_t)k0 * LHW;
    a.x  = ap[0];      a.y  = ap[LHW];
    const float* bp0 = bptr0 + (size_t)k0 * RDIM;
    b0.x = bp0[0];     b0.y = bp0[RDIM];
    const float* bp1 = bptr1 + (size_t)k0 * RDIM;
    b1.x = bp1[0];     b1.y = bp1[RDIM];
    // D = A(16x4) * B(4x16) + C, fp32 throughout (matches fp32 reference)
    acc0 = __builtin_amdgcn_wmma_f32_16x16x4_f32(
        /*neg_a=*/false, a, /*neg_b=*/false, b0,
        /*c_mod=*/(short)0, acc0, /*reuse_a=*/false, /*reuse_b=*/false);
    acc1 = __builtin_amdgcn_wmma_f32_16x16x4_f32(
        false, a, false, b1, (short)0, acc1, false, false);
  }

  const int rc = rate_choice[g];
  #pragma unroll
  for (int ct = 0; ct < 2; ++ct) {
    const int r  = col0 + ct * 16 + lrow;          // output column
    const float mv = (r < rc) ? 1.0f : 0.0f;
    const float bv = bias[g * RDIM + r];
    #pragma unroll
    for (int v = 0; v < 8; ++v) {
      const int M = v + half * 8;                  // D row in this lane's VGPR v
      if (M < nvalid) {
        const int t = s_tok[M];
        const int b = t >> 8;
        const int l = t & (LHW - 1);
        const size_t off = (size_t)b * (RDIM * LHW) + (size_t)r * LHW + l;
        const float d = ct ? acc1[v] : acc0[v];
        outY[off] = (d + bv) * mv;
        outM[off] = mv;
      }
    }
  }
}

// ---------------------------------------------------------------------------
// Host-side launcher
// Inputs (setup_inputs order): x[f32], indexes[int], weight[f32], bias[f32],
// rate_choice[int32]. Output: x_masked (1048576 f32) then mask (1048576 f32).
// ---------------------------------------------------------------------------
extern "C" void kernel_launch(void* const* d_in, const int* in_sizes, int n_in,
                              void* d_out, int out_size, void* d_ws, size_t ws_size,
                              hipStream_t stream) {
  const float* x      = (const float*)d_in[0];
  const int*   idx    = (const int*)d_in[1];
  const float* weight = (const float*)d_in[2];
  const float* bias   = (const float*)d_in[3];
  const int*   rc     = (const int*)d_in[4];

  float* outY = (float*)d_out;
  float* outM = outY + (size_t)16 * RDIM * LHW;   // second tuple element

  // Workspace: 8 counts (aligned to 64B) + 8*4096 u16 token lists (~64.03 KB)
  int* counts = (int*)d_ws;
  unsigned short* lists = (unsigned short*)((char*)d_ws + 64);

  build_group_lists<<<dim3(1), dim3(256), 0, stream>>>(idx, counts, lists);
  rate_adaption_wmma<<<dim3(NGRP * (NTOK / 16)), dim3(256), 0, stream>>>(
      x, weight, bias, rc, counts, lists, outY, outM);
}